// PythonRenderer_10685878632928
// MI455X (gfx1250) — compile-verified
//
#include <hip/hip_runtime.h>

// -----------------------------------------------------------------------------
// Deferred rasterizer interpolation pass for MI455X (gfx1250).
// Bandwidth/latency-bound gather kernel: no matrix structure -> no WMMA.
//  - wave32, 1 pixel/lane, 131072 waves for latency hiding
//  - gather tables (~15MB) stay resident in the 192MB L2
//  - streaming traffic (~168MB: index read + 4 output images) uses NT hints
//  - pointer-based addressing so clang merges row gathers into single
//    global_load_b96/b64 requests and merges the NT stores (17 vmem/pixel,
//    verified in round-1 histogram; unsigned scale_offset form scalarized
//    them into 33 vmem/pixel and was reverted)
// -----------------------------------------------------------------------------

namespace {
constexpr int kB = 4;
constexpr int kH = 1024;
constexpr int kW = 1024;
constexpr int kV = 98304;
constexpr int kN = kB * kH * kW;          // 4,194,304 pixels
constexpr float kEps = 1e-8f;
}

__device__ __forceinline__ float epsclamp(float v) {
  // where(x < 0, min(x, -EPS), max(x, EPS))
  return v < 0.0f ? fminf(v, -kEps) : fmaxf(v, kEps);
}

__global__ __launch_bounds__(256, 8) void raster_interp_kernel(
    const float* __restrict__ v2d,        // [B,V,3] screen-space verts
    const float* __restrict__ vt,         // [T,2]   uv table
    const int*   __restrict__ vi,         // [F,3]   vertex ids per face
    const int*   __restrict__ vti,        // [F,3]   uv ids per face
    const int*   __restrict__ index_img,  // [B,H,W] face id per pixel (-1 = bg)
    const float* __restrict__ vn,         // [B,V,3] normals
    float* __restrict__ depth_o,          // [B,H,W]
    float* __restrict__ bary_o,           // [B,H,W,3]
    float* __restrict__ vt_o,             // [B,H,W,2]
    float* __restrict__ vn_o)             // [B,H,W,3]
{
  const int pix = blockIdx.x * 256 + threadIdx.x;
  if (pix >= kN) return;

  const int b = pix >> 20;                // H*W = 2^20
  const int y = (pix >> 10) & (kH - 1);   // row  (pcy)
  const int x = pix & (kW - 1);           // col  (pcx)

  // Streaming read-once: non-temporal so it does not pollute L2.
  const int fidx = __builtin_nontemporal_load(index_img + pix);
  const float fm = (fidx != -1) ? 1.0f : 0.0f;
  const int ii = (fidx < 0) ? 0 : fidx;

  // Face tables: 12-byte rows, merged into one global_load_b96 each.
  const int* __restrict__ vr = vi + 3 * ii;
  const int* __restrict__ tr = vti + 3 * ii;
  const int i0 = vr[0], i1 = vr[1], i2 = vr[2];
  const int t0i = tr[0], t1i = tr[1], t2i = tr[2];

  // Vertex gathers (12B rows -> b96).
  const float* __restrict__ vb = v2d + b * (3 * kV);
  const float* q0 = vb + 3 * i0;
  const float* q1 = vb + 3 * i1;
  const float* q2 = vb + 3 * i2;
  const float p0x = q0[0], p0y = q0[1], p0z = q0[2];
  const float p1x = q1[0], p1y = q1[1], p1z = q1[2];
  const float p2x = q2[0], p2y = q2[1], p2z = q2[2];

  // UV gathers (8B rows -> b64).
  const float* u0 = vt + 2 * t0i;
  const float* u1 = vt + 2 * t1i;
  const float* u2 = vt + 2 * t2i;
  const float t0x = u0[0], t0y = u0[1];
  const float t1x = u1[0], t1y = u1[1];
  const float t2x = u2[0], t2y = u2[1];

  // Normal gathers (12B rows -> b96).
  const float* __restrict__ nb = vn + b * (3 * kV);
  const float* m0 = nb + 3 * i0;
  const float* m1 = nb + 3 * i1;
  const float* m2 = nb + 3 * i2;
  const float n0x = m0[0], n0y = m0[1], n0z = m0[2];
  const float n1x = m1[0], n1y = m1[1], n1z = m1[2];
  const float n2x = m2[0], n2y = m2[1], n2z = m2[2];

  // ---- barycentrics --------------------------------------------------------
  const float e1x = p1x - p0x, e1y = p1y - p0y;   // vec01
  const float e2x = p2x - p0x, e2y = p2y - p0y;   // vec02
  const float det = e1x * e2y - e1y * e2x;
  const float denom = epsclamp(det);

  const float px = (float)x - p0x;
  const float py = (float)y - p0y;
  const float l1 = (px * e2y - py * e2x) / denom;
  const float l2 = (py * e1x - px * e1y) / denom;
  const float l0 = 1.0f - l1 - l2;

  // ---- perspective correction ---------------------------------------------
  const float w0 = 1.0f / epsclamp(p0z);
  const float w1 = 1.0f / epsclamp(p1z);
  const float w2 = 1.0f / epsclamp(p2z);
  const float zi = 1.0f / epsclamp(w0 * l0 + w1 * l1 + w2 * l2);
  const float b0 = w0 * l0 * zi;
  const float b1 = w1 * l1 * zi;
  const float b2 = w2 * l2 * zi;

  // ---- outputs (write-once streaming -> non-temporal stores; adjacent
  //      scalar NT stores merge into b96/b64 NT stores) ----------------------
  __builtin_nontemporal_store(zi * fm, depth_o + pix);

  const int bo = 3 * pix;
  __builtin_nontemporal_store(b0 * fm, bary_o + bo + 0);
  __builtin_nontemporal_store(b1 * fm, bary_o + bo + 1);
  __builtin_nontemporal_store(b2 * fm, bary_o + bo + 2);

  // uv interpolation + NDC-lerp against pixel-center base
  const float uu = t0x * b0 + t1x * b1 + t2x * b2;
  const float vv = t0y * b0 + t1y * b1 + t2y * b2;
  const float basex = (float)(2 * x + 1) * (1.0f / (float)kW) - 1.0f; // exact /2^10
  const float basey = (float)(2 * y + 1) * (1.0f / (float)kH) - 1.0f;
  const float vtx = basex + fm * ((uu * 2.0f - 1.0f) - basex);
  const float vty = basey + fm * ((vv * 2.0f - 1.0f) - basey);
  const int to = 2 * pix;
  __builtin_nontemporal_store(vtx, vt_o + to + 0);
  __builtin_nontemporal_store(vty, vt_o + to + 1);

  // normals: (n . b) * fm   (fm in {0,1} => identical to reference's fm^2 form)
  const float vnx = (n0x * b0 + n1x * b1 + n2x * b2) * fm;
  const float vny = (n0y * b0 + n1y * b1 + n2y * b2) * fm;
  const float vnz = (n0z * b0 + n1z * b1 + n2z * b2) * fm;
  __builtin_nontemporal_store(vnx, vn_o + bo + 0);
  __builtin_nontemporal_store(vny, vn_o + bo + 1);
  __builtin_nontemporal_store(vnz, vn_o + bo + 2);
}

extern "C" void kernel_launch(void* const* d_in, const int* in_sizes, int n_in,
                              void* d_out, int out_size, void* d_ws, size_t ws_size,
                              hipStream_t stream) {
  (void)in_sizes; (void)n_in; (void)out_size; (void)d_ws; (void)ws_size;

  // setup_inputs() order: v2d, vt, vi, vti, index_img, vn
  const float* v2d       = (const float*)d_in[0];
  const float* vt        = (const float*)d_in[1];
  const int*   vi        = (const int*)  d_in[2];
  const int*   vti       = (const int*)  d_in[3];
  const int*   index_img = (const int*)  d_in[4];
  const float* vn        = (const float*)d_in[5];

  // Outputs concatenated flat in return order:
  // depth [N], bary [N,3], vt_img [N,2], vn_img [N,3]  (N = B*H*W)
  float* out     = (float*)d_out;
  float* depth_o = out;
  float* bary_o  = out + (size_t)kN;
  float* vt_o    = out + (size_t)4 * kN;
  float* vn_o    = out + (size_t)6 * kN;

  const int threads = 256;
  const int blocks = (kN + threads - 1) / threads;  // 16384
  raster_interp_kernel<<<blocks, threads, 0, stream>>>(
      v2d, vt, vi, vti, index_img, vn, depth_o, bary_o, vt_o, vn_o);
}